// Net_DirFiletr_57088705298886
// MI455X (gfx1250) — compile-verified
//
#include <hip/hip_runtime.h>

typedef float v2f __attribute__((ext_vector_type(2)));
typedef float v8f __attribute__((ext_vector_type(8)));

#define RR    7
#define TAPS  8
#define TILE  16
#define HALO  30      // TILE + 2*RR
#define LDSW  33      // padded row width (float2 units) to stagger LDS banks
#define KCH   6       // K = 24 handled as 6 chunks of 4
#define IMGW  1024
#define IMGH  1024

// Band(row, k) = 1 iff row <= k <= row+7  (selects the 8 taps incl. center)
static __device__ __forceinline__ float bandv(int row, int k) {
  return (k >= row && k <= row + RR) ? 1.0f : 0.0f;
}

// Horizontal directional sums: D = X(16x24) * Band(24x16).
// co = 0 -> dir (0,-1) taps at halo cols [j..j+7]; co = 7 -> dir (0,+1) taps [j+7..j+14].
static __device__ __forceinline__ void wmma_h(const float2 (*T)[LDSW], int lane, int co,
                                              const v2f* bnd, float* x1, float* y2) {
  const int hi = lane >> 4, m = lane & 15;
  v8f a1 = {0.f, 0.f, 0.f, 0.f, 0.f, 0.f, 0.f, 0.f};
  v8f a2 = {0.f, 0.f, 0.f, 0.f, 0.f, 0.f, 0.f, 0.f};
#pragma unroll
  for (int q = 0; q < KCH; ++q) {
    const int kx = 4 * q + 2 * hi;                 // A-matrix K layout: VGPR0 = K0/K2, VGPR1 = K1/K3
    const int c0 = min(co + kx,     HALO - 1);     // k=23 has band weight 0; clamp keeps data finite
    const int c1 = min(co + kx + 1, HALO - 1);
    const float2 p0 = T[m + RR][c0];
    const float2 p1 = T[m + RR][c1];
    v2f ax; ax.x = p0.x; ax.y = p1.x;              // x stream
    v2f ay; ay.x = p0.y; ay.y = p1.y;              // x^2 stream
    a1 = __builtin_amdgcn_wmma_f32_16x16x4_f32(false, ax, false, bnd[q], (short)0, a1, false, false);
    a2 = __builtin_amdgcn_wmma_f32_16x16x4_f32(false, ay, false, bnd[q], (short)0, a2, false, false);
  }
#pragma unroll
  for (int v = 0; v < 8; ++v) { x1[v] = a1[v]; y2[v] = a2[v]; }
}

// Vertical directional sums: D = Band(16x24) * X(24x16).
// ro = 0 -> dir (-1,0) taps at halo rows [i..i+7]; ro = 7 -> dir (+1,0) taps [i+7..i+14].
static __device__ __forceinline__ void wmma_v(const float2 (*T)[LDSW], int lane, int ro,
                                              const v2f* bnd, float* x1, float* y2) {
  const int hi = lane >> 4, n = lane & 15;
  v8f a1 = {0.f, 0.f, 0.f, 0.f, 0.f, 0.f, 0.f, 0.f};
  v8f a2 = {0.f, 0.f, 0.f, 0.f, 0.f, 0.f, 0.f, 0.f};
#pragma unroll
  for (int q = 0; q < KCH; ++q) {
    const int kx = 4 * q + 2 * hi;
    const int r0 = min(ro + kx,     HALO - 1);
    const int r1 = min(ro + kx + 1, HALO - 1);
    const float2 p0 = T[r0][n + RR];
    const float2 p1 = T[r1][n + RR];
    v2f bx; bx.x = p0.x; bx.y = p1.x;
    v2f by; by.x = p0.y; by.y = p1.y;
    a1 = __builtin_amdgcn_wmma_f32_16x16x4_f32(false, bnd[q], false, bx, (short)0, a1, false, false);
    a2 = __builtin_amdgcn_wmma_f32_16x16x4_f32(false, bnd[q], false, by, (short)0, a2, false, false);
  }
#pragma unroll
  for (int v = 0; v < 8; ++v) { x1[v] = a1[v]; y2[v] = a2[v]; }
}

// Diagonal directional sums (not matmul-structured): direct VALU from LDS.
static __device__ __forceinline__ void diag_sum(const float2 (*T)[LDSW], int lane,
                                                int dy, int dx, float* x1, float* y2) {
  const int hi = lane >> 4, n = lane & 15;
#pragma unroll
  for (int v = 0; v < 8; ++v) {
    const int r = v + 8 * hi + RR;   // same pixel mapping as the WMMA C/D layout
    const int c = n + RR;
    float s1 = 0.f, s2 = 0.f;
#pragma unroll
    for (int t = 0; t < TAPS; ++t) {
      const float2 p = T[r + dy * t][c + dx * t];
      s1 += p.x; s2 += p.y;
    }
    x1[v] = s1; y2[v] = s2;
  }
}

// Running argmin update; key = y2 - x1^2/8 is a monotone proxy for the variance.
static __device__ __forceinline__ void upd(float* bestKey, float* bestX1,
                                           const float* x1, const float* y2) {
#pragma unroll
  for (int v = 0; v < 8; ++v) {
    const float key = fmaf(-0.125f * x1[v], x1[v], y2[v]);
    if (key < bestKey[v]) { bestKey[v] = key; bestX1[v] = x1[v]; }  // strict < keeps first-min
  }
}

__global__ __launch_bounds__(256)
void dirfilter_kernel(const float* __restrict__ xin_all, float* __restrict__ out_all) {
  __shared__ float2 tile[8][HALO][LDSW];            // 63,360 B: one halo tile per wave

  const int lane = threadIdx.x;                     // 0..31 (wave32)
  const int w    = threadIdx.y;                     // wave id 0..7
  float2 (*T)[LDSW] = tile[w];

  const int tx  = blockIdx.x * 8 + w;               // tile column 0..63
  const int ty  = blockIdx.y;                       // tile row    0..63
  const int img = blockIdx.z;                       // image       0..7
  const int gx0 = tx * TILE - RR;
  const int gy0 = ty * TILE - RR;
  const float* __restrict__ xin = xin_all + (size_t)img * IMGW * IMGH;

  // Stage 30x30 halo as {x, x^2} pairs; zero-pad outside the image (matches conv zero padding).
  for (int idx = lane; idx < HALO * HALO; idx += 32) {
    const int r = idx / HALO, c = idx - r * HALO;
    const int gy = gy0 + r, gx = gx0 + c;
    float v = 0.0f;
    if ((unsigned)gy < (unsigned)IMGH && (unsigned)gx < (unsigned)IMGW)
      v = xin[gy * IMGW + gx];
    T[r][c] = make_float2(v, v * v);
  }
  __syncthreads();

  // Band operand chunks (shared by all 4 axis directions and both quantities).
  const int hi = lane >> 4, n = lane & 15;
  v2f bnd[KCH];
#pragma unroll
  for (int q = 0; q < KCH; ++q) {
    const int kx = 4 * q + 2 * hi;
    bnd[q].x = bandv(n, kx);
    bnd[q].y = bandv(n, kx + 1);
  }

  float bestKey[8], bestX1[8], x1[8], y2[8];

  // Reference direction order: (-1,-1) (-1,0) (-1,1) (0,-1) (0,1) (1,-1) (1,0) (1,1)
  diag_sum(T, lane, -1, -1, x1, y2);                // dir 0 initializes the running best
#pragma unroll
  for (int v = 0; v < 8; ++v) {
    bestKey[v] = fmaf(-0.125f * x1[v], x1[v], y2[v]);
    bestX1[v]  = x1[v];
  }
  wmma_v (T, lane, 0,      bnd, x1, y2); upd(bestKey, bestX1, x1, y2);  // (-1, 0)
  diag_sum(T, lane, -1, +1,      x1, y2); upd(bestKey, bestX1, x1, y2); // (-1,+1)
  wmma_h (T, lane, 0,      bnd, x1, y2); upd(bestKey, bestX1, x1, y2);  // ( 0,-1)
  wmma_h (T, lane, RR,     bnd, x1, y2); upd(bestKey, bestX1, x1, y2);  // ( 0,+1)
  diag_sum(T, lane, +1, -1,      x1, y2); upd(bestKey, bestX1, x1, y2); // (+1,-1)
  wmma_v (T, lane, RR,     bnd, x1, y2); upd(bestKey, bestX1, x1, y2);  // (+1, 0)
  diag_sum(T, lane, +1, +1,      x1, y2); upd(bestKey, bestX1, x1, y2); // (+1,+1)

  // Write the selected directional mean (x1/8).
  float* __restrict__ o = out_all + (size_t)img * IMGW * IMGH;
#pragma unroll
  for (int v = 0; v < 8; ++v) {
    const int i = v + 8 * hi;
    o[(ty * TILE + i) * IMGW + tx * TILE + n] = bestX1[v] * 0.125f;
  }
}

extern "C" void kernel_launch(void* const* d_in, const int* in_sizes, int n_in,
                              void* d_out, int out_size, void* d_ws, size_t ws_size,
                              hipStream_t stream) {
  (void)in_sizes; (void)n_in; (void)d_ws; (void)ws_size; (void)out_size;
  const float* x = (const float*)d_in[0];   // [8,1,1024,1024] f32 (d_in[1] = fixed dir kernels, R=7 baked in)
  float* out = (float*)d_out;               // [8,1,1024,1024] f32

  dim3 block(32, 8);                        // 8 wave32 waves, one 16x16 tile each
  dim3 grid(IMGW / (TILE * 8), IMGH / TILE, 8);
  dirfilter_kernel<<<grid, block, 0, stream>>>(x, out);
}